// MultiStepQuantumModel_25769804416
// MI455X (gfx1250) — compile-verified
//
#include <hip/hip_runtime.h>

// MultiStepQuantumModel — closed form:
//   gammas = sigmoid(raw_gammas)                       [64]
//   P      = prod(1 - gammas)                          scalar
//   expvals[i] = cos(thetas[i]) * P + (1 - P)          [B]
// Output layout (tuple flattened): [expvals(B), gammas(64)].
// phis is mathematically dead -> never loaded (saves 64 MiB/call of traffic).
//
// Memory-bound: 128 MiB/call / 23.3 TB/s ~= 5.8 us floor. Strategy:
// b128 vectorized streaming, x2-unrolled grid-stride loop (2 independent
// loads in flight per wave), lookahead global_prefetch_b8, wave32-sized
// blocks. 128 MiB working set fits the 192 MB L2 across graph replays, so
// default RT temporal hints are kept (NT/LU would evict re-used lines).

#define QM_BLOCK 256
#define QM_GRID  4096

// ---------------------------------------------------------------------------
// Kernel 1: tiny — sigmoid of the 64 raw_gammas, survival product P.
// One 64-thread block (2 waves). Writes gammas to d_out tail, P to d_ws.
// ---------------------------------------------------------------------------
__global__ __launch_bounds__(64) void qm_gamma_kernel(
    const float* __restrict__ raw_gammas,
    float* __restrict__ gammas_out,   // d_out + B
    float* __restrict__ surv_out,     // d_ws[0]
    int n_steps) {
  __shared__ float g[64];
  const int t = threadIdx.x;

  float one_minus = 1.0f;               // neutral element for the product
  if (t < n_steps) {
    const float x   = raw_gammas[t];
    // sigmoid(x) = 1/(1+e^-x); v_rcp_f32 accuracy (~1 ulp) is plenty here.
    const float sig = __builtin_amdgcn_rcpf(1.0f + __expf(-x));
    gammas_out[t]   = sig;
    one_minus       = 1.0f - sig;
  }
  g[t] = one_minus;
  __syncthreads();

  // LDS tree product over 64 entries
  for (int off = 32; off > 0; off >>= 1) {
    if (t < off) g[t] *= g[t + off];
    __syncthreads();
  }
  if (t == 0) surv_out[0] = g[0];
}

// ---------------------------------------------------------------------------
// Kernel 2: streaming — expvals[i] = cos(thetas[i]) * P + (1 - P).
// float4 (global_load_b128/global_store_b128). Grid-stride loop unrolled x2:
// both b128 loads issue before either result is consumed, doubling per-wave
// outstanding memory. Prefetch runs one super-stride ahead (speculative:
// OOB prefetch addresses are silently dropped by the translation path).
// ---------------------------------------------------------------------------
__global__ __launch_bounds__(QM_BLOCK) void qm_expval_kernel(
    const float4* __restrict__ thetas4,
    float4* __restrict__ out4,
    const float* __restrict__ surv_p,
    int n4,
    const float* __restrict__ thetas_tail,
    float* __restrict__ out_tail,
    int tail) {
  const float s = *surv_p;        // uniform scalar, L2-resident broadcast
  const float b = 1.0f - s;

  const int stride = (int)(gridDim.x * blockDim.x);
  int i = (int)(blockIdx.x * blockDim.x + threadIdx.x);

  // Main x2-unrolled stream: pairs (i, i+stride), super-stride 2*stride.
  for (; i + stride < n4; i += 2 * stride) {
    __builtin_prefetch(&thetas4[i + 2 * stride], 0, 1);
    __builtin_prefetch(&thetas4[i + 3 * stride], 0, 1);

    const float4 ta = thetas4[i];            // load #1 in flight
    const float4 tb = thetas4[i + stride];   // load #2 in flight

    float4 ra, rb;
    ra.x = __cosf(ta.x) * s + b;
    ra.y = __cosf(ta.y) * s + b;
    ra.z = __cosf(ta.z) * s + b;
    ra.w = __cosf(ta.w) * s + b;
    rb.x = __cosf(tb.x) * s + b;
    rb.y = __cosf(tb.y) * s + b;
    rb.z = __cosf(tb.z) * s + b;
    rb.w = __cosf(tb.w) * s + b;

    out4[i]          = ra;
    out4[i + stride] = rb;
  }

  // Odd leftover stride-chunk.
  if (i < n4) {
    const float4 t = thetas4[i];
    float4 r;
    r.x = __cosf(t.x) * s + b;
    r.y = __cosf(t.y) * s + b;
    r.z = __cosf(t.z) * s + b;
    r.w = __cosf(t.w) * s + b;
    out4[i] = r;
  }

  // Scalar tail (B % 4 elements) — empty for B = 16M, kept for generality.
  if (blockIdx.x == 0) {
    const int t = threadIdx.x;
    if (t < tail) {
      out_tail[t] = __cosf(thetas_tail[t]) * s + b;
    }
  }
}

// ---------------------------------------------------------------------------
// Launch wrapper
// ---------------------------------------------------------------------------
extern "C" void kernel_launch(void* const* d_in, const int* in_sizes, int n_in,
                              void* d_out, int out_size, void* d_ws, size_t ws_size,
                              hipStream_t stream) {
  const float* thetas     = (const float*)d_in[0];
  // d_in[1] = phis: mathematically irrelevant to <Z>, never read.
  const float* raw_gammas = (const float*)d_in[2];

  const int B       = in_sizes[0];
  const int n_steps = in_sizes[2];

  float* expvals    = (float*)d_out;
  float* gammas_out = (float*)d_out + B;
  float* surv       = (float*)d_ws;   // 1 float of scratch

  // 1) 64 sigmoids + survival product P -> d_ws (same stream => ordered).
  qm_gamma_kernel<<<1, 64, 0, stream>>>(raw_gammas, gammas_out, surv, n_steps);

  // 2) Bandwidth-bound streaming map.
  const int n4   = B >> 2;
  const int tail = B & 3;
  int grid = QM_GRID;
  const int needed = (n4 + QM_BLOCK - 1) / QM_BLOCK;
  if (needed < grid) grid = (needed > 0) ? needed : 1;

  qm_expval_kernel<<<grid, QM_BLOCK, 0, stream>>>(
      (const float4*)thetas, (float4*)expvals, surv, n4,
      thetas + ((size_t)n4 << 2), expvals + ((size_t)n4 << 2), tail);
}